// Seq2Seq_11793980195010
// MI455X (gfx1250) — compile-verified
//
#include <hip/hip_runtime.h>
#include <cstdint>
#include <cstddef>

// Problem constants (match reference)
#define TT   1024
#define BB   512
#define VV   29
#define HH   64
#define G4   256     // 4*H gate width
#define SOSI 27

#define NWG     16   // workgroups; each owns ROWS batch rows
#define ROWS    32
#define THREADS 256  // 8 wave32 waves

typedef float v2f __attribute__((ext_vector_type(2)));
typedef float v8f __attribute__((ext_vector_type(8)));

// fp32 WMMA: D[16x16] = A[16x4] * B[4x16] + C   (8-arg VOP3P form)
#define WMMA_F32X4(a, b, c) \
  __builtin_amdgcn_wmma_f32_16x16x4_f32(false, (a), false, (b), (short)0, (c), false, false)

__device__ __forceinline__ float sigm(float x) { return 1.0f / (1.0f + __expf(-x)); }
__device__ __forceinline__ float tanhx(float x) { return 1.0f - 2.0f / (__expf(2.0f * x) + 1.0f); }

// G[32x256] = A[32xKTOT] @ B[KTOT x 256]   (all LDS)
// A columns [0,KSPLIT) come from A0 (stride ldA0), [KSPLIT,KTOT) from A1 (stride ldA1).
// 8 waves: wave w computes N-tiles {2w,2w+1} x M-tiles {0,1}. EXEC all-ones.
template <int KTOT, int KSPLIT>
__device__ __forceinline__ void gemm32x256(const float* A0, int ldA0,
                                           const float* A1, int ldA1,
                                           const float* B, int ldB,
                                           float* G, int ldG)
{
  const int tid   = threadIdx.x;
  const int wave  = tid >> 5;
  const int lane  = tid & 31;
  const int l16   = lane & 15;
  const int lhalf = lane >> 4;        // 0 or 1
  const int khalf = lhalf << 1;       // 0 or 2
  const int n0 = (wave * 2) * 16;
  const int n1 = n0 + 16;

  v8f acc00 = {}; v8f acc01 = {}; v8f acc10 = {}; v8f acc11 = {};

#pragma unroll
  for (int kk = 0; kk < KTOT; kk += 4) {
    // KSPLIT is a multiple of 4, so kk decides the A segment for the whole wave.
    const float* Ap = (kk < KSPLIT) ? A0 : A1;
    const int    la = (kk < KSPLIT) ? ldA0 : ldA1;
    const int    ac = ((kk < KSPLIT) ? kk : (kk - KSPLIT)) + khalf;
    const int    k0 = kk + khalf;

    v2f af0, af1, bf0, bf1;
    af0[0] = Ap[l16 * la + ac];        af0[1] = Ap[l16 * la + ac + 1];
    af1[0] = Ap[(16 + l16) * la + ac]; af1[1] = Ap[(16 + l16) * la + ac + 1];
    bf0[0] = B[k0 * ldB + n0 + l16];   bf0[1] = B[(k0 + 1) * ldB + n0 + l16];
    bf1[0] = B[k0 * ldB + n1 + l16];   bf1[1] = B[(k0 + 1) * ldB + n1 + l16];

    acc00 = WMMA_F32X4(af0, bf0, acc00);
    acc01 = WMMA_F32X4(af0, bf1, acc01);
    acc10 = WMMA_F32X4(af1, bf0, acc10);
    acc11 = WMMA_F32X4(af1, bf1, acc11);
  }

  const int rbase = lhalf * 8;
#pragma unroll
  for (int r = 0; r < 8; ++r) {
    const int row0 = rbase + r;        // M-tile 0
    const int row1 = 16 + rbase + r;   // M-tile 1
    G[row0 * ldG + n0 + l16] = acc00[r];
    G[row0 * ldG + n1 + l16] = acc01[r];
    G[row1 * ldG + n0 + l16] = acc10[r];
    G[row1 * ldG + n1 + l16] = acc11[r];
  }
}

// LSTM gate nonlinearity: gates in G (+bias bb per gate column), state C, h -> H[:, hoff..]
__device__ __forceinline__ void lstm_act(const float* G, int ldG, const float* bb,
                                         float* C, float* H, int ldH, int hoff)
{
  const int tid = threadIdx.x;
#pragma unroll
  for (int e = 0; e < 8; ++e) {
    const int idx = e * THREADS + tid;          // 32*64 elements
    const int r = idx >> 6, j = idx & 63;
    const float ig = sigm (G[r * ldG + j]        + bb[j]);
    const float fg = sigm (G[r * ldG + 64 + j]   + bb[64 + j]);
    const float gg = tanhx(G[r * ldG + 128 + j]  + bb[128 + j]);
    const float og = sigm (G[r * ldG + 192 + j]  + bb[192 + j]);
    const float c  = fg * C[r * 64 + j] + ig * gg;
    C[r * 64 + j] = c;
    H[r * ldH + hoff + j] = og * tanhx(c);
  }
}

// ---------------- sync init ----------------
__global__ void init_sync_kernel(unsigned* __restrict__ gmask, unsigned* __restrict__ gcnt)
{
  const int i = blockIdx.x * blockDim.x + threadIdx.x;
  if (i < TT) { gmask[i] = 0u; gcnt[i] = 0u; }
}

// ---------------- fused 2-layer encoder ----------------
// LDS floats: W0 96*264 + W1 128*264 + X0 32*36 + XH1 32*132 + G 32*260 + C0,C1 2*2048 + b 512
#define ENC_SMEM_FLOATS (96*264 + 128*264 + 32*36 + 32*132 + 32*260 + 2*32*64 + 512)

__global__ __launch_bounds__(THREADS) void enc_kernel(
    const float* __restrict__ in,
    const float* __restrict__ Wih0, const float* __restrict__ Whh0,
    const float* __restrict__ bih0, const float* __restrict__ bhh0,
    const float* __restrict__ Wih1, const float* __restrict__ Whh1,
    const float* __restrict__ bih1, const float* __restrict__ bhh1,
    float* __restrict__ hfin, float* __restrict__ cfin)
{
  extern __shared__ float sm[];
  float* W0  = sm;                 // [96][264]  rows 0..31: Wih0^T (29 valid), 32..95: Whh0^T
  float* W1  = W0  + 96 * 264;     // [128][264] rows 0..63: Wih1^T, 64..127: Whh1^T
  float* X0  = W1  + 128 * 264;    // [32][36]   x_t padded to 32 cols
  float* XH1 = X0  + 32 * 36;      // [32][132]  cols 0..63 h0, 64..127 h1
  float* G   = XH1 + 32 * 132;     // [32][260]
  float* C0  = G   + 32 * 260;     // [32][64]
  float* C1  = C0  + 32 * 64;
  float* bl0 = C1  + 32 * 64;      // [256]
  float* bl1 = bl0 + 256;

  const int tid = threadIdx.x;
  const int b0row = blockIdx.x * ROWS;

  // Stage weights/bias once.
  for (int s = tid; s < 96 * 256; s += THREADS) {
    const int k = s >> 8, n = s & 255;
    W0[k * 264 + n] = (k < 32) ? ((k < VV) ? Wih0[n * VV + k] : 0.0f)
                               : Whh0[n * HH + (k - 32)];
  }
  for (int s = tid; s < 128 * 256; s += THREADS) {
    const int k = s >> 8, n = s & 255;
    W1[k * 264 + n] = (k < 64) ? Wih1[n * HH + k] : Whh1[n * HH + (k - 64)];
  }
  bl0[tid] = bih0[tid] + bhh0[tid];
  bl1[tid] = bih1[tid] + bhh1[tid];
  for (int s = tid; s < 32 * 132; s += THREADS) XH1[s] = 0.0f;
  for (int s = tid; s < 32 * 36;  s += THREADS) X0[s] = 0.0f;
  for (int s = tid; s < 32 * 64;  s += THREADS) { C0[s] = 0.0f; C1[s] = 0.0f; }
  __syncthreads();

  for (int t = 0; t < TT; ++t) {
    // load x_t tile (29 cols, zero padded to 32)
    for (int s = tid; s < 32 * 32; s += THREADS) {
      const int r = s >> 5, v = s & 31;
      X0[r * 36 + v] = (v < VV)
          ? in[(size_t)t * BB * VV + (size_t)(b0row + r) * VV + v] : 0.0f;
    }
    // prefetch next timestep's slice (one 64B line per thread)
    if (t + 1 < TT && tid < 58)
      __builtin_prefetch(in + (size_t)(t + 1) * BB * VV + (size_t)b0row * VV + tid * 16, 0, 1);
    __syncthreads();

    // layer 0: K = 32 (x) + 64 (h0)
    gemm32x256<96, 32>(X0, 36, XH1, 132, W0, 264, G, 260);
    __syncthreads();
    lstm_act(G, 260, bl0, C0, XH1, 132, 0);   // h0_new -> XH1[:,0:64]
    __syncthreads();

    // layer 1: K = 64 (h0_new) + 64 (h1)
    gemm32x256<128, 128>(XH1, 132, XH1, 132, W1, 264, G, 260);
    __syncthreads();
    lstm_act(G, 260, bl1, C1, XH1, 132, 64);  // h1_new -> XH1[:,64:128]
    __syncthreads();
  }

  // final top-layer (h, c) -> workspace for decoder
  for (int s = tid; s < 32 * 64; s += THREADS) {
    const int r = s >> 6, j = s & 63;
    hfin[(size_t)(b0row + r) * HH + j] = XH1[r * 132 + 64 + j];
    cfin[(size_t)(b0row + r) * HH + j] = C1[r * 64 + j];
  }
}

// ---------------- decoder ----------------
#define DEC_SMEM_FLOATS (64*264 + 128*264 + 64*40 + 256*32 + 32*132 + 32*260 + 2*32*64 + 256 + 256 + 256 + 32 + 4)

__global__ __launch_bounds__(THREADS) void dec_kernel(
    const float* __restrict__ d1Wih, const float* __restrict__ d1Whh,
    const float* __restrict__ d1bih, const float* __restrict__ d1bhh,
    const float* __restrict__ d2Wih, const float* __restrict__ d2Whh,
    const float* __restrict__ d2bih, const float* __restrict__ d2bhh,
    const float* __restrict__ clsW, const float* __restrict__ clsb,
    const float* __restrict__ hfin, const float* __restrict__ cfin,
    unsigned* __restrict__ gmask, unsigned* __restrict__ gcnt,
    float* __restrict__ out)
{
  extern __shared__ float sm[];
  float* W1d  = sm;                  // [64][264]   d1_Whh^T
  float* W2d  = W1d + 64 * 264;      // [128][264]  [d2_Wih^T ; d2_Whh^T]
  float* Wc   = W2d + 128 * 264;     // [64][40]    cls_W^T, cols padded to 32(+8)
  float* WihD = Wc  + 64 * 40;       // [256][32]   d1_Wih, per-thread row layout
  float* XHd  = WihD + 256 * 32;     // [32][132]   cols 0..63 h1, 64..127 h2
  float* G    = XHd + 32 * 132;      // [32][260]   gates / pred (reused)
  float* C1d  = G   + 32 * 260;      // [32][64]
  float* C2d  = C1d + 32 * 64;
  float* xw   = C2d + 32 * 64;       // [256] per-step d1 input+bias vector
  float* b2d  = xw  + 256;           // [256]
  float* b1s  = b2d + 256;           // [256] d1_bih+d1_bhh
  float* bc   = b1s + 256;           // [32]
  unsigned* um = (unsigned*)(bc + 32);  // um[0]=local mask, um[1]=shared mask

  const int tid   = threadIdx.x;
  const int wave  = tid >> 5;
  const int lane  = tid & 31;
  const int l16   = lane & 15;
  const int lhalf = lane >> 4;
  const int khalf = lhalf << 1;
  const int b0row = blockIdx.x * ROWS;

  // Stage weights.
  for (int s = tid; s < 64 * 256; s += THREADS) {
    const int k = s >> 8, n = s & 255;
    W1d[k * 264 + n] = d1Whh[n * HH + k];
  }
  for (int s = tid; s < 128 * 256; s += THREADS) {
    const int k = s >> 8, n = s & 255;
    W2d[k * 264 + n] = (k < 64) ? d2Wih[n * HH + k] : d2Whh[n * HH + (k - 64)];
  }
  for (int s = tid; s < 64 * 32; s += THREADS) {
    const int k = s >> 5, v = s & 31;
    Wc[k * 40 + v] = (v < VV) ? clsW[v * HH + k] : 0.0f;
  }
  for (int s = tid; s < 256 * 32; s += THREADS) {
    const int n = s >> 5, v = s & 31;
    WihD[n * 32 + v] = (v < VV) ? d1Wih[n * VV + v] : 0.0f;
  }
  b2d[tid] = d2bih[tid] + d2bhh[tid];
  b1s[tid] = d1bih[tid] + d1bhh[tid];
  if (tid < 32) bc[tid] = (tid < VV) ? clsb[tid] : 0.0f;

  // State init: h1,c1 from encoder; h2=c2=0.
  for (int s = tid; s < 32 * 64; s += THREADS) {
    const int r = s >> 6, j = s & 63;
    XHd[r * 132 + j]      = hfin[(size_t)(b0row + r) * HH + j];
    XHd[r * 132 + 64 + j] = 0.0f;
    C1d[s] = cfin[(size_t)(b0row + r) * HH + j];
    C2d[s] = 0.0f;
  }
  __syncthreads();

  unsigned mask = 1u << SOSI;   // let0 one-hot at SOS

  for (int t = 0; t < TT; ++t) {
    // xw[n] = d1 biases + sum of Wih columns in the multi-hot mask (let entries are 1.0)
    {
      float s = b1s[tid];
      unsigned m = mask;
      while (m) { const int v = __builtin_ctz(m); m &= m - 1; s += WihD[tid * 32 + v]; }
      xw[tid] = s;
    }
    if (tid == 0) um[0] = 0u;
    __syncthreads();

    // d1: K=64 over h1
    gemm32x256<64, 64>(XHd, 132, XHd, 132, W1d, 264, G, 260);
    __syncthreads();
    lstm_act(G, 260, xw, C1d, XHd, 132, 0);   // h1_new -> XHd[:,0:64]
    __syncthreads();

    // d2: K=128 over [h1_new | h2]
    gemm32x256<128, 128>(XHd, 132, XHd, 132, W2d, 264, G, 260);
    __syncthreads();
    lstm_act(G, 260, b2d, C2d, XHd, 132, 64); // h2_new -> XHd[:,64:128]
    __syncthreads();

    // classifier: pred[32,29] = h2 @ clsW^T + bc   (waves 0,1; 2x2 WMMA tiles)
    if (wave < 2) {
      const float* A = XHd + 64;               // h2 region, stride 132
      const int n0 = wave * 16;
      v8f acc0 = {}; v8f acc1 = {};
#pragma unroll
      for (int kk = 0; kk < 64; kk += 4) {
        const int k0 = kk + khalf;
        v2f af0, af1, bf0;
        af0[0] = A[l16 * 132 + k0];        af0[1] = A[l16 * 132 + k0 + 1];
        af1[0] = A[(16 + l16) * 132 + k0]; af1[1] = A[(16 + l16) * 132 + k0 + 1];
        bf0[0] = Wc[k0 * 40 + n0 + l16];   bf0[1] = Wc[(k0 + 1) * 40 + n0 + l16];
        acc0 = WMMA_F32X4(af0, bf0, acc0);
        acc1 = WMMA_F32X4(af1, bf0, acc1);
      }
      const int rbase = lhalf * 8;
      const int col = n0 + l16;
      const float bcv = bc[col];
      float* outp = out + (size_t)t * BB * VV + (size_t)b0row * VV;
#pragma unroll
      for (int r = 0; r < 8; ++r) {
        const int row0 = rbase + r, row1 = 16 + rbase + r;
        const float p0 = acc0[r] + bcv, p1 = acc1[r] + bcv;
        G[row0 * 260 + col] = p0;
        G[row1 * 260 + col] = p1;
        if (col < VV) {
          outp[(size_t)row0 * VV + col] = p0;
          outp[(size_t)row1 * VV + col] = p1;
        }
      }
    }
    __syncthreads();

    // per-row argmax (first-max, like jnp.argmax) -> local mask
    if (tid < 32) {
      const float* p = G + tid * 260;
      int best = 0; float bv = p[0];
      for (int v = 1; v < VV; ++v) { const float pv = p[v]; if (pv > bv) { bv = pv; best = v; } }
      atomicOr(&um[0], 1u << best);
    }
    __syncthreads();

    // device-wide union of masks + phase barrier (16 WGs, L2 atomics)
    if (tid == 0) {
      __hip_atomic_fetch_or(&gmask[t], um[0], __ATOMIC_RELAXED, __HIP_MEMORY_SCOPE_AGENT);
      __hip_atomic_fetch_add(&gcnt[t], 1u, __ATOMIC_RELEASE, __HIP_MEMORY_SCOPE_AGENT);
      while (__hip_atomic_load(&gcnt[t], __ATOMIC_ACQUIRE, __HIP_MEMORY_SCOPE_AGENT) < (unsigned)NWG)
        __builtin_amdgcn_s_sleep(1);
      um[1] = __hip_atomic_load(&gmask[t], __ATOMIC_RELAXED, __HIP_MEMORY_SCOPE_AGENT);
    }
    __syncthreads();
    mask = um[1];
  }
}

// ---------------- launcher ----------------
extern "C" void kernel_launch(void* const* d_in, const int* in_sizes, int n_in,
                              void* d_out, int out_size, void* d_ws, size_t ws_size,
                              hipStream_t stream)
{
  const float* in    = (const float*)d_in[0];
  const float* eWih0 = (const float*)d_in[1];
  const float* eWhh0 = (const float*)d_in[2];
  const float* ebih0 = (const float*)d_in[3];
  const float* ebhh0 = (const float*)d_in[4];
  const float* eWih1 = (const float*)d_in[5];
  const float* eWhh1 = (const float*)d_in[6];
  const float* ebih1 = (const float*)d_in[7];
  const float* ebhh1 = (const float*)d_in[8];
  const float* d1Wih = (const float*)d_in[9];
  const float* d1Whh = (const float*)d_in[10];
  const float* d1bih = (const float*)d_in[11];
  const float* d1bhh = (const float*)d_in[12];
  const float* d2Wih = (const float*)d_in[13];
  const float* d2Whh = (const float*)d_in[14];
  const float* d2bih = (const float*)d_in[15];
  const float* d2bhh = (const float*)d_in[16];
  const float* clsW  = (const float*)d_in[17];
  const float* clsb  = (const float*)d_in[18];
  float* out = (float*)d_out;

  // workspace: [mask:1024 u32][cnt:1024 u32][hfin:512*64 f32][cfin:512*64 f32]
  unsigned* gmask = (unsigned*)d_ws;
  unsigned* gcnt  = gmask + TT;
  float* hfin = (float*)(gcnt + TT);
  float* cfin = hfin + (size_t)BB * HH;

  init_sync_kernel<<<(TT + 255) / 256, 256, 0, stream>>>(gmask, gcnt);

  enc_kernel<<<NWG, THREADS, ENC_SMEM_FLOATS * sizeof(float), stream>>>(
      in, eWih0, eWhh0, ebih0, ebhh0, eWih1, eWhh1, ebih1, ebhh1, hfin, cfin);

  dec_kernel<<<NWG, THREADS, DEC_SMEM_FLOATS * sizeof(float), stream>>>(
      d1Wih, d1Whh, d1bih, d1bhh, d2Wih, d2Whh, d2bih, d2bhh,
      clsW, clsb, hfin, cfin, gmask, gcnt, out);
}